// tensor_train_completion_395136991290
// MI455X (gfx1250) — compile-verified
//
#include <hip/hip_runtime.h>
#include <hip/hip_bf16.h>

typedef __attribute__((ext_vector_type(2))) float v2f;
typedef __attribute__((ext_vector_type(8))) float v8f;

#define RANK   16
#define NDIM   320
#define NCOL   (NDIM * RANK)          // 5120 = n1 * r2
#define NTILEN (NCOL / 16)            // 320 tiles along N
#define NTILEM (NDIM / 16)            // 20 tiles along M
#define WAVES_PER_BLOCK 8             // 256 threads = 8 wave32

// ---------------------------------------------------------------------------
// Stage A: H(320 x 5120) = G0(320 x 16) @ G1(16 x 5120), pure f32 WMMA.
// One wave per 16x16 output tile; K=16 done as 4 chained V_WMMA_F32_16X16X4_F32.
// Grid is sized to cover tiles exactly (6400 waves = 800 blocks) so EXEC is
// all-1s with no divergence around the WMMA ops (ISA 7.12 requirement).
// ---------------------------------------------------------------------------
__global__ __launch_bounds__(256) void tt_stageA_wmma(const float* __restrict__ g0,
                                                      const float* __restrict__ g1,
                                                      float* __restrict__ H) {
    const int lane = threadIdx.x & 31;
    const int wave = threadIdx.x >> 5;
    const int t    = blockIdx.x * WAVES_PER_BLOCK + wave;   // tile id 0..6399
    const int m0   = (t / NTILEN) * 16;                     // row tile base in G0/H
    const int n0   = (t % NTILEN) * 16;                     // col tile base in G1/H
    const int half = lane >> 4;                             // 0: K{0,1}, 1: K{2,3}
    const int l    = lane & 15;

    v8f c = {0.f, 0.f, 0.f, 0.f, 0.f, 0.f, 0.f, 0.f};
#pragma unroll
    for (int k = 0; k < 4; ++k) {
        // A operand (16x4): lane supplies A[m0+l, 4k + 2*half + {0,1}]
        const float* ap = g0 + (m0 + l) * RANK + 4 * k + 2 * half;
        v2f a;
        a.x = ap[0];
        a.y = ap[1];
        // B operand (4x16): lane supplies B[4k + 2*half + {0,1}, n0+l]
        const float* bp = g1 + (size_t)(4 * k + 2 * half) * NCOL + n0 + l;
        v2f b;
        b.x = bp[0];
        b.y = bp[NCOL];
        // 8 args: (neg_a, A, neg_b, B, c_mod, C, reuse_a, reuse_b)
        c = __builtin_amdgcn_wmma_f32_16x16x4_f32(false, a, false, b,
                                                  (short)0, c, false, false);
    }
    // D layout: VGPR r holds C[m0 + r + 8*half, n0 + l]
    float* out = H + (size_t)(m0 + 8 * half) * NCOL + n0 + l;
#pragma unroll
    for (int r = 0; r < 8; ++r) out[(size_t)r * NCOL] = c[r];
}

// ---------------------------------------------------------------------------
// Stage B: gather + length-16 dot against g2 column. H rows are 64B-aligned
// (i1*16 floats), so read as 4x float4. All traffic is L2-resident (6.55MB).
// ---------------------------------------------------------------------------
__global__ __launch_bounds__(256) void tt_stageB_gather(const float* __restrict__ H,
                                                        const float* __restrict__ g2,
                                                        const int* __restrict__ idxs,
                                                        float* __restrict__ out,
                                                        int N) {
    const int s = blockIdx.x * blockDim.x + threadIdx.x;
    if (s >= N) return;
    const int i0 = idxs[3 * s + 0];
    const int i1 = idxs[3 * s + 1];
    const int i2 = idxs[3 * s + 2];
    const float4* h = (const float4*)(H + (size_t)i0 * NCOL + (size_t)i1 * RANK);
    float acc = 0.f;
#pragma unroll
    for (int q = 0; q < 4; ++q) {
        const float4 hv = h[q];
        acc += hv.x * g2[(4 * q + 0) * NDIM + i2];
        acc += hv.y * g2[(4 * q + 1) * NDIM + i2];
        acc += hv.z * g2[(4 * q + 2) * NDIM + i2];
        acc += hv.w * g2[(4 * q + 3) * NDIM + i2];
    }
    out[s] = acc;
}

// ---------------------------------------------------------------------------
// Fallback (only if ws_size can't hold H): direct per-sample bilinear form.
// 256 FMA + ~1.2KB of L2-resident gathered reads per sample.
// ---------------------------------------------------------------------------
__global__ __launch_bounds__(256) void tt_direct(const float* __restrict__ g0,
                                                 const float* __restrict__ g1,
                                                 const float* __restrict__ g2,
                                                 const int* __restrict__ idxs,
                                                 float* __restrict__ out,
                                                 int N) {
    const int s = blockIdx.x * blockDim.x + threadIdx.x;
    if (s >= N) return;
    const int i0 = idxs[3 * s + 0];
    const int i1 = idxs[3 * s + 1];
    const int i2 = idxs[3 * s + 2];

    float w[RANK];
#pragma unroll
    for (int b = 0; b < RANK; ++b) w[b] = g2[b * NDIM + i2];

    float acc = 0.f;
#pragma unroll
    for (int a = 0; a < RANK; ++a) {
        const float va = g0[i0 * RANK + a];
        const float4* m = (const float4*)(g1 + (size_t)a * NCOL + (size_t)i1 * RANK);
        float dot = 0.f;
#pragma unroll
        for (int q = 0; q < 4; ++q) {
            const float4 mv = m[q];
            dot += mv.x * w[4 * q + 0];
            dot += mv.y * w[4 * q + 1];
            dot += mv.z * w[4 * q + 2];
            dot += mv.w * w[4 * q + 3];
        }
        acc += va * dot;
    }
    out[s] = acc;
}

extern "C" void kernel_launch(void* const* d_in, const int* in_sizes, int n_in,
                              void* d_out, int out_size, void* d_ws, size_t ws_size,
                              hipStream_t stream) {
    const float* g0   = (const float*)d_in[0];   // (1, 320, 16)
    const float* g1   = (const float*)d_in[1];   // (16, 320, 16) == (16, 5120)
    const float* g2   = (const float*)d_in[2];   // (16, 320, 1)
    const int*   idxs = (const int*)d_in[3];     // (N, 3)
    float*       out  = (float*)d_out;
    const int    N    = in_sizes[3] / 3;

    const size_t needH = (size_t)NDIM * NCOL * sizeof(float);  // 6,553,600 B

    if (ws_size >= needH) {
        float* H = (float*)d_ws;
        // 20*320 = 6400 tiles, 8 waves/block -> 800 blocks, exact coverage.
        tt_stageA_wmma<<<(NTILEM * NTILEN) / WAVES_PER_BLOCK, 256, 0, stream>>>(g0, g1, H);
        tt_stageB_gather<<<(N + 255) / 256, 256, 0, stream>>>(H, g2, idxs, out, N);
    } else {
        tt_direct<<<(N + 255) / 256, 256, 0, stream>>>(g0, g1, g2, idxs, out, N);
    }
}